// QuantumRBF_65481071403320
// MI455X (gfx1250) — compile-verified
//
#include <hip/hip_runtime.h>

// QuantumRBF fidelity kernel for MI455X (gfx1250, wave32).
//
// fidelity[b] = prod_{w=0..7} cos^2( (DEPTH*0.5) * (x[b,w] - y[b,w]) )
// DEPTH = 2  ->  half-angle scale = 1.0
//
// Memory-bound: 268 MB read + 17 MB write @ 23.3 TB/s => ~12 us floor.
// Strategy: one row (8 floats) per thread, 2x b128 NT loads per input,
// native v_cos_f32 for the transcendental, NT store, and gfx1250
// global_prefetch_b8 to deepen memory-level parallelism.

typedef float f32x4 __attribute__((ext_vector_type(4)));

#define HALF_SCALE 1.0f   // DEPTH(=2) * 0.5
#define BLOCK 256         // 8 wave32 waves per workgroup

__global__ __launch_bounds__(BLOCK) void quantum_rbf_fidelity(
    const float* __restrict__ x,
    const float* __restrict__ y,
    float* __restrict__ out,
    int n_rows)
{
    const int row = blockIdx.x * BLOCK + threadIdx.x;
    if (row >= n_rows) return;

    const f32x4* __restrict__ xp = (const f32x4*)(x + (size_t)row * 8);
    const f32x4* __restrict__ yp = (const f32x4*)(y + (size_t)row * 8);

    // Speculative prefetch ~64KB ahead (gfx1250 global_prefetch_b8).
    // Translation failures past the end of the buffer are silently dropped.
    __builtin_prefetch((const char*)xp + 65536, 0, 0);
    __builtin_prefetch((const char*)yp + 65536, 0, 0);

    // Streaming 128-bit non-temporal loads: no reuse, keep L2 clean.
    f32x4 x0 = __builtin_nontemporal_load(xp);
    f32x4 x1 = __builtin_nontemporal_load(xp + 1);
    f32x4 y0 = __builtin_nontemporal_load(yp);
    f32x4 y1 = __builtin_nontemporal_load(yp + 1);

    f32x4 d0 = (x0 - y0) * HALF_SCALE;
    f32x4 d1 = (x1 - y1) * HALF_SCALE;

    // Product of cos^2 over the 8 wires. Native v_cos_f32 is plenty
    // accurate for |d| ~ few units (inputs are N(0,1) differences).
    float p0 = 1.0f, p1 = 1.0f;
#pragma unroll
    for (int i = 0; i < 4; ++i) {
        float c0 = __cosf(d0[i]);
        float c1 = __cosf(d1[i]);
        p0 *= c0 * c0;
        p1 *= c1 * c1;
    }

    __builtin_nontemporal_store(p0 * p1, out + row);
}

extern "C" void kernel_launch(void* const* d_in, const int* in_sizes, int n_in,
                              void* d_out, int out_size, void* d_ws, size_t ws_size,
                              hipStream_t stream)
{
    (void)n_in; (void)d_ws; (void)ws_size;

    const float* x = (const float*)d_in[0];   // [B, 8] f32
    const float* y = (const float*)d_in[1];   // [B, 8] f32
    float* out     = (float*)d_out;           // [B, 1] f32

    const int n_rows = out_size;              // B = in_sizes[0] / 8
    (void)in_sizes;

    const int grid = (n_rows + BLOCK - 1) / BLOCK;
    quantum_rbf_fidelity<<<grid, BLOCK, 0, stream>>>(x, y, out, n_rows);
}